// VelocityLoss_84576495993037
// MI455X (gfx1250) — compile-verified
//
#include <hip/hip_runtime.h>
#include <hip/hip_bf16.h>

// Problem constants (from reference)
#define Bn 512
#define Sn 480
#define Fn 149
#define NCH 30   // chunks of 16 s-values covering s = 1..480 (s=480 masked off)

typedef __attribute__((ext_vector_type(2))) float v2f;
typedef __attribute__((ext_vector_type(8))) float v8f;

// W_MAT (24 x 5) padded to (24 x 16) with zeros so B-fragment loads are
// unconditional (no exec-masked branches). Values derived exactly as numpy:
// wsum = [8,10,10,10,10]; W[j, parts[j]] = weight[j]/wsum[parts[j]]
#define Z11 0.f,0.f,0.f,0.f,0.f,0.f,0.f,0.f,0.f,0.f,0.f
__constant__ float cWpad[24][16] = {
    {0.f, 1.f/10.f, 0.f, 0.f, 0.f, Z11},
    {0.f, 2.f/10.f, 0.f, 0.f, 0.f, Z11},
    {0.f, 3.f/10.f, 0.f, 0.f, 0.f, Z11},
    {0.f, 4.f/10.f, 0.f, 0.f, 0.f, Z11},
    {0.f, 0.f, 1.f/10.f, 0.f, 0.f, Z11},
    {0.f, 0.f, 2.f/10.f, 0.f, 0.f, Z11},
    {0.f, 0.f, 3.f/10.f, 0.f, 0.f, Z11},
    {0.f, 0.f, 4.f/10.f, 0.f, 0.f, Z11},
    {1.f/8.f, 0.f, 0.f, 0.f, 0.f, Z11},
    {1.f/8.f, 0.f, 0.f, 0.f, 0.f, Z11},
    {1.f/8.f, 0.f, 0.f, 0.f, 0.f, Z11},
    {1.f/8.f, 0.f, 0.f, 0.f, 0.f, Z11},
    {1.f/8.f, 0.f, 0.f, 0.f, 0.f, Z11},
    {0.f, 0.f, 0.f, 1.f/10.f, 0.f, Z11},
    {0.f, 0.f, 0.f, 2.f/10.f, 0.f, Z11},
    {0.f, 0.f, 0.f, 3.f/10.f, 0.f, Z11},
    {0.f, 0.f, 0.f, 4.f/10.f, 0.f, Z11},
    {0.f, 0.f, 0.f, 0.f, 1.f/10.f, Z11},
    {0.f, 0.f, 0.f, 0.f, 2.f/10.f, Z11},
    {0.f, 0.f, 0.f, 0.f, 3.f/10.f, Z11},
    {0.f, 0.f, 0.f, 0.f, 4.f/10.f, Z11},
    {1.f/8.f, 0.f, 0.f, 0.f, 0.f, Z11},
    {2.f/8.f, 0.f, 0.f, 0.f, 0.f, Z11},
    {1.f/8.f, 0.f, 0.f, 0.f, 0.f, Z11},
};
#undef Z11

__global__ void init_acc(double* acc) {
    if (threadIdx.x < 3) acc[threadIdx.x] = 0.0;
}

// Main kernel: one wave per (b, chunk-of-16-s). accA -> acc[0], accC -> acc[1].
__global__ __launch_bounds__(256) void vel_main(
        const float* __restrict__ pred,   // (B,S,F)
        const float* __restrict__ tvf,    // (B,S+1,VF)
        const float* __restrict__ mean_,  // (F,)
        const float* __restrict__ std_,   // (F,)
        double* __restrict__ acc)
{
    __shared__ float lnorm[8][16 * 24];   // per-wave 16(M=s) x 24(K=joint) norms
    __shared__ float redA[8], redC[8];

    const int tid  = threadIdx.x;
    const int wv   = tid >> 5;
    const int lane = tid & 31;
    const int g    = blockIdx.x * 8 + wv;     // 0 .. 15359
    const int b    = g / NCH;
    const int ch   = g % NCH;
    const int s0   = 1 + 16 * ch;             // first s of chunk

    const int  mlo  = lane & 15;
    const bool hilf = (lane >= 16);
    const int  koff = hilf ? 2 : 0;

    // ---- hoisted B fragments (chunk-invariant, unconditional loads) ----
    // B 4x16 f32 layout: lane&15 = N; VGPR0 = row K(4c4+koff), VGPR1 = row K(4c4+koff+1)
    float bfx[6], bfy[6];
    #pragma unroll
    for (int c4 = 0; c4 < 6; ++c4) {
        const int k0 = 4 * c4 + koff;
        bfx[c4] = cWpad[k0][mlo];
        bfy[c4] = cWpad[k0 + 1][mlo];
    }

    // ---- per-lane normalization params (lanes 0..23 own joint L = lane) ----
    const int L = lane;
    float stdp0 = 1.f, stdp1 = 1.f, stdp2 = 1.f;
    float mv0 = 0.f, mv1 = 0.f, mv2 = 0.f;
    float sv0 = 1.f, sv1 = 1.f, sv2 = 1.f;
    if (L < 24) {
        stdp0 = std_[3 * L + 0]; stdp1 = std_[3 * L + 1]; stdp2 = std_[3 * L + 2];
        mv0 = mean_[72 + 3 * L + 0]; mv1 = mean_[72 + 3 * L + 1]; mv2 = mean_[72 + 3 * L + 2];
        sv0 = std_[72 + 3 * L + 0]; sv1 = std_[72 + 3 * L + 1]; sv2 = std_[72 + 3 * L + 2];
    }

    float accA = 0.f;

    // ---- phase 1: velocities, velocity-MSE, norms -> LDS ----
    float prev0 = 0.f, prev1 = 0.f, prev2 = 0.f;
    if (L < 24) {
        const float* row = pred + ((size_t)b * Sn + (s0 - 1)) * Fn + 3 * L;
        prev0 = row[0]; prev1 = row[1]; prev2 = row[2];
    }
    #pragma unroll 4
    for (int i = 0; i < 16; ++i) {
        const int s = s0 + i;                 // wave-uniform
        if (s < Sn) {
            if (L < 24) {
                const float* rowp = pred + ((size_t)b * Sn + s) * Fn + 3 * L;
                const float* rowv = pred + ((size_t)b * Sn + s) * Fn + 77 + 3 * L;
                float c0 = rowp[0], c1 = rowp[1], c2 = rowp[2];
                float r0 = rowv[0], r1 = rowv[1], r2 = rowv[2];
                float v0 = (c0 - prev0) * stdp0;
                float v1 = (c1 - prev1) * stdp1;
                float v2 = (c2 - prev2) * stdp2;
                lnorm[wv][i * 24 + L] = sqrtf(v0 * v0 + v1 * v1 + v2 * v2);
                float d0 = r0 - (v0 - mv0) / sv0;
                float d1 = r1 - (v1 - mv1) / sv1;
                float d2 = r2 - (v2 - mv2) / sv2;
                accA += d0 * d0 + d1 * d1 + d2 * d2;
                prev0 = c0; prev1 = c1; prev2 = c2;
            }
        } else {
            if (L < 24) lnorm[wv][i * 24 + L] = 0.f;
        }
    }

    __syncthreads();  // publish per-wave norm tiles (uniform control flow here)

    // ---- phase 2: factor = norms(16x24) @ W(24x5) via 6 chained WMMA f32 16x16x4 ----
    float fac[8];
#if __has_builtin(__builtin_amdgcn_wmma_f32_16x16x4_f32)
    v8f accm = {0.f, 0.f, 0.f, 0.f, 0.f, 0.f, 0.f, 0.f};
    #pragma unroll
    for (int c4 = 0; c4 < 6; ++c4) {
        const int k0 = 4 * c4 + koff;
        v2f Af, Bf;
        // A 16x4 f32 layout: lane&15 = M; VGPR0 = K(k0), VGPR1 = K(k0+1)
        Af.x = lnorm[wv][mlo * 24 + k0];
        Af.y = lnorm[wv][mlo * 24 + k0 + 1];
        Bf.x = bfx[c4];
        Bf.y = bfy[c4];
        accm = __builtin_amdgcn_wmma_f32_16x16x4_f32(
            false, Af, false, Bf, (short)0, accm, false, false);
    }
    #pragma unroll
    for (int r = 0; r < 8; ++r) fac[r] = accm[r];
#else
    // Scalar fallback: fac[r] = D[M = r (+8 for hi half)][N = mlo]
    #pragma unroll
    for (int r = 0; r < 8; ++r) {
        const int M = r + (hilf ? 8 : 0);
        float f = 0.f;
        for (int k = 0; k < 24; ++k) f += lnorm[wv][M * 24 + k] * cWpad[k][mlo];
        fac[r] = f;
    }
#endif

    // ---- phase 3: factor-MSE vs _true_vel_factor ----
    float accC = 0.f;
    if (mlo < 5) {
        const float mf = mean_[144 + mlo];
        const float sf = std_[144 + mlo];
        #pragma unroll
        for (int r = 0; r < 8; ++r) {
            const int s = s0 + r + (hilf ? 8 : 0);   // D row M maps to s0 + M
            if (s < Sn) {
                float t = tvf[((size_t)b * (Sn + 1) + s) * 5 + mlo];
                float f = (fac[r] - mf) / sf;
                float e = t - f;
                accC += e * e;
            }
        }
    }

    // ---- reduction: wave shuffle -> LDS -> double atomics ----
    #pragma unroll
    for (int o = 16; o > 0; o >>= 1) {
        accA += __shfl_xor(accA, o, 32);
        accC += __shfl_xor(accC, o, 32);
    }
    if (lane == 0) { redA[wv] = accA; redC[wv] = accC; }
    __syncthreads();
    if (tid == 0) {
        float a = 0.f, c = 0.f;
        #pragma unroll
        for (int w = 0; w < 8; ++w) { a += redA[w]; c += redC[w]; }
        atomicAdd(&acc[0], (double)a);
        atomicAdd(&acc[1], (double)c);
    }
}

// Sum of pred_vel[:,0,:]^2  ->  acc[2]
__global__ __launch_bounds__(256) void vel_zero(const float* __restrict__ pred,
                                                double* __restrict__ acc)
{
    __shared__ float red[8];
    const int idx  = blockIdx.x * 256 + threadIdx.x;   // 0 .. 36863
    const int b    = idx / 72;
    const int j    = idx % 72;
    const int lane = threadIdx.x & 31;
    const int wv   = threadIdx.x >> 5;

    float v = pred[((size_t)b * Sn) * Fn + 77 + j];
    float z = v * v;
    #pragma unroll
    for (int o = 16; o > 0; o >>= 1) z += __shfl_xor(z, o, 32);
    if (lane == 0) red[wv] = z;
    __syncthreads();
    if (threadIdx.x == 0) {
        float s = 0.f;
        #pragma unroll
        for (int w = 0; w < 8; ++w) s += red[w];
        atomicAdd(&acc[2], (double)s);
    }
}

__global__ void vel_finalize(const double* __restrict__ acc, float* __restrict__ out)
{
    const double A = acc[0];           // sum over B*(S-1)*72 of (pred_vel - _train_velocity)^2
    const double C = acc[1];           // sum over B*(S-1)*5  of factor error^2
    const double Z = acc[2];           // sum over B*72 of pred_vel[:,0]^2
    const double n1 = (double)Bn * (Sn - 1) * 72.0;
    const double nz = (double)Bn * 72.0;
    const double n2 = (double)Bn * (Sn - 1) * 5.0;
    const double loss1 = (A / n1) * 10.0 + (Z / nz) * 20.0;
    const double loss2 = (C / n2) * 10.0;
    out[0] = (float)(loss1 * 2.0 + loss2 * 1.5);
}

extern "C" void kernel_launch(void* const* d_in, const int* in_sizes, int n_in,
                              void* d_out, int out_size, void* d_ws, size_t ws_size,
                              hipStream_t stream)
{
    (void)in_sizes; (void)n_in; (void)out_size; (void)ws_size;
    const float* pred  = (const float*)d_in[0];   // predict_seq (B,S,F)
    // d_in[1] (_train_x1) and d_in[2] (_train_x2) are dead: s=0 terms are sliced away.
    const float* tvf   = (const float*)d_in[3];   // _true_vel_factor (B,S+1,VF)
    const float* mean_ = (const float*)d_in[4];   // (F,)
    const float* std_  = (const float*)d_in[5];   // (F,)
    double* acc = (double*)d_ws;                  // [A, C, Z]
    float* out  = (float*)d_out;

    init_acc<<<1, 32, 0, stream>>>(acc);
    vel_main<<<(Bn * NCH) / 8, 256, 0, stream>>>(pred, tvf, mean_, std_, acc);
    vel_zero<<<(Bn * 72) / 256, 256, 0, stream>>>(pred, acc);
    vel_finalize<<<1, 1, 0, stream>>>(acc, out);
}